// PPOAgent_38319698215283
// MI455X (gfx1250) — compile-verified
//
#include <hip/hip_runtime.h>
#include <math.h>

typedef __attribute__((ext_vector_type(2))) float v2f;
typedef __attribute__((ext_vector_type(8))) float v8f;

#define T_TOT   131072
#define D_DIM   512
#define A_DIM   17
#define GAMMA_  0.99f
#define LAMDA_  0.9f
#define CLAMP_  0.3f
#define ENTC_   0.05f
#define VALC_   0.5f

// ---- GAE scan config ----
#define NCHUNK 128
#define CTH    256
#define SUB    (T_TOT / (NCHUNK * CTH))     // 4 elements per thread

// ---- main fused kernel config ----
#define G5_BLOCKS      128
#define G5_THREADS     256
#define WAVES_PER_BLK  (G5_THREADS / 32)    // 8 (wave32)
#define TOT_WAVES      (G5_BLOCKS * WAVES_PER_BLK)  // 1024
#define NTILES         (T_TOT / 16)         // 8192
#define TILE_ITERS     (NTILES / TOT_WAVES) // 8, exact

#define WTS 516   // padded k-stride of W' in LDS (bank-conflict-free b64)
#define STG 33    // padded row stride of epilogue stage tile

// ------------------------------------------------------------------
__global__ void init_acc(float* acc) {
    if (threadIdx.x < 16) acc[threadIdx.x] = 0.0f;
}

// ------------------------------------------------------------------
// K1: per-chunk affine composition (reverse):  x_start = A + B * x_end
__global__ void gae_partials(const float* __restrict__ rewards,
                             const float* __restrict__ values,
                             const int*   __restrict__ term,
                             float* __restrict__ chA, float* __restrict__ chB) {
    __shared__ float sa[CTH], sb[CTH];
    const int blk = blockIdx.x, tid = threadIdx.x;
    const int base = blk * (CTH * SUB) + tid * SUB;
    float a = 0.0f, b = 1.0f;
    for (int i = SUB - 1; i >= 0; --i) {
        int t = base + i;
        float mk = term[t] ? 0.0f : 1.0f;
        float nv = (t + 1 < T_TOT) ? values[t + 1] : 0.0f;
        float d  = rewards[t] + GAMMA_ * nv * mk - values[t];
        float c  = GAMMA_ * LAMDA_ * mk;
        a = d + c * a;
        b = c * b;
    }
    sa[tid] = a; sb[tid] = b;
    __syncthreads();
    if (tid == 0) {
        float A = 0.0f, B = 1.0f;
        for (int j = CTH - 1; j >= 0; --j) { A = sa[j] + sb[j] * A; B = sb[j] * B; }
        chA[blk] = A; chB[blk] = B;
    }
}

// K2: serial scan over 128 chunk compositions (reverse) -> carry into each chunk
__global__ void gae_carry(const float* __restrict__ chA, const float* __restrict__ chB,
                          float* __restrict__ carry) {
    if (threadIdx.x == 0 && blockIdx.x == 0) {
        float g = 0.0f;
        for (int c = NCHUNK - 1; c >= 0; --c) { carry[c] = g; g = chA[c] + chB[c] * g; }
    }
}

// K3: replay chunks with carries; write adv/ret; accumulate sum & sumsq
__global__ void gae_apply(const float* __restrict__ rewards,
                          const float* __restrict__ values,
                          const int*   __restrict__ term,
                          const float* __restrict__ carry,
                          float* __restrict__ adv, float* __restrict__ ret,
                          float* __restrict__ acc) {
    __shared__ float sa[CTH], sb[CTH], sc[CTH], red[CTH];
    const int blk = blockIdx.x, tid = threadIdx.x;
    const int base = blk * (CTH * SUB) + tid * SUB;

    float dloc[SUB], cloc[SUB];
    float a = 0.0f, b = 1.0f;
    for (int i = SUB - 1; i >= 0; --i) {
        int t = base + i;
        float mk = term[t] ? 0.0f : 1.0f;
        float nv = (t + 1 < T_TOT) ? values[t + 1] : 0.0f;
        float d  = rewards[t] + GAMMA_ * nv * mk - values[t];
        float c  = GAMMA_ * LAMDA_ * mk;
        dloc[i] = d; cloc[i] = c;
        a = d + c * a;
        b = c * b;
    }
    sa[tid] = a; sb[tid] = b;
    __syncthreads();
    if (tid == 0) {
        float g = carry[blk];
        for (int j = CTH - 1; j >= 0; --j) { sc[j] = g; g = sa[j] + sb[j] * g; }
    }
    __syncthreads();

    float g = sc[tid], s = 0.0f, s2 = 0.0f;
    for (int i = SUB - 1; i >= 0; --i) {
        int t = base + i;
        g = dloc[i] + cloc[i] * g;
        adv[t] = g;
        ret[t] = g + values[t];
        s += g; s2 += g * g;
    }
    red[tid] = s; __syncthreads();
    for (int off = CTH / 2; off > 0; off >>= 1) {
        if (tid < off) red[tid] += red[tid + off];
        __syncthreads();
    }
    if (tid == 0) atomicAdd(&acc[0], red[0]);
    __syncthreads();
    red[tid] = s2; __syncthreads();
    for (int off = CTH / 2; off > 0; off >>= 1) {
        if (tid < off) red[tid] += red[tid + off];
        __syncthreads();
    }
    if (tid == 0) atomicAdd(&acc[1], red[0]);
}

// K4: mean / 1/(std(ddof=1)+1e-5)
__global__ void gae_stats(float* acc) {
    if (threadIdx.x == 0 && blockIdx.x == 0) {
        float S = acc[0], S2 = acc[1];
        float mean = S / (float)T_TOT;
        float var  = (S2 - S * mean) / (float)(T_TOT - 1);
        var = fmaxf(var, 0.0f);
        acc[5] = mean;
        acc[6] = 1.0f / (sqrtf(var) + 1e-5f);
    }
}

// ------------------------------------------------------------------
// K5: fused WMMA GEMM ([T,512] x [512,32'] where col0..16=W, col17=w_v)
//     + per-row log-softmax + PPO loss partial sums.
__global__ void __launch_bounds__(G5_THREADS)
ppo_main(const float* __restrict__ states,
         const int*   __restrict__ actions,
         const float* __restrict__ old_log_probs,
         const float* __restrict__ W,
         const float* __restrict__ w_v,
         const float* __restrict__ adv,
         const float* __restrict__ ret,
         float* __restrict__ acc) {
    extern __shared__ float smem[];
    float* Wt    = smem;                          // 32 * WTS floats (transposed, padded)
    float* stage = Wt + 32 * WTS;                 // WAVES_PER_BLK * 16 * STG
    float* red   = stage + WAVES_PER_BLK * 16 * STG; // G5_THREADS

    const int tid  = threadIdx.x;
    const int lane = tid & 31;
    const int warp = tid >> 5;

    // Cooperative fill of W' transposed: Wt[n][k]; n<17 -> W, n==17 -> w_v, else 0
    for (int idx = tid; idx < 32 * D_DIM; idx += G5_THREADS) {
        int n = idx >> 9;         // /512
        int k = idx & (D_DIM - 1);
        float v = 0.0f;
        if (n < A_DIM)      v = W[k * A_DIM + n];
        else if (n == A_DIM) v = w_v[k];
        Wt[n * WTS + k] = v;
    }
    __syncthreads();

    const float meanv = acc[5];
    const float rstd  = acc[6];

    // fp32 WMMA lane mapping (16x16x4_f32):
    // A: lane m = lane&15, K pair base = (lane>>4)*2, v2f = A[m][kb], A[m][kb+1]
    // B: lane n = lane&15, same K split, v2f = W'[n][kb], W'[n][kb+1]
    const int m    = lane & 15;
    const int koff = (lane >> 4) << 1;
    const float* wt0 = Wt + (size_t)m * WTS + koff;            // N-tile 0 (cols 0..15)
    const float* wt1 = Wt + (size_t)(m + 16) * WTS + koff;     // N-tile 1 (cols 16..31)
    float* stageW = stage + warp * 16 * STG;
    const int rowsel = (lane < 16) ? 0 : 8;

    const int gw = blockIdx.x * WAVES_PER_BLK + warp;

    float pl = 0.0f, vl = 0.0f, ent = 0.0f;

    for (int it = 0; it < TILE_ITERS; ++it) {
        const int tile = gw + it * TOT_WAVES;
        const float* aBase = states + (size_t)tile * 16 * D_DIM + (size_t)m * D_DIM + koff;

        v8f c0 = {};
        v8f c1 = {};
#pragma unroll 4
        for (int kk = 0; kk < D_DIM; kk += 4) {
            v2f a  = __builtin_nontemporal_load((const v2f*)(aBase + kk)); // streams; states > L2
            v2f b0 = *(const v2f*)(wt0 + kk);
            v2f b1 = *(const v2f*)(wt1 + kk);
            c0 = __builtin_amdgcn_wmma_f32_16x16x4_f32(false, a, false, b0,
                                                       (short)0, c0, false, false);
            c1 = __builtin_amdgcn_wmma_f32_16x16x4_f32(false, a, false, b1,
                                                       (short)0, c1, false, false);
        }

        // Stage the 16x32 tile: C layout VGPR r -> row r (lanes0-15) / r+8 (lanes16-31)
#pragma unroll
        for (int r = 0; r < 8; ++r) {
            stageW[(r + rowsel) * STG + m]      = c0[r];
            stageW[(r + rowsel) * STG + 16 + m] = c1[r];
        }
        __syncthreads();   // uniform: every wave runs exactly TILE_ITERS iterations

        if (lane < 16) {
            const int t = tile * 16 + lane;
            const float* row = stageW + lane * STG;
            float mx = row[0];
#pragma unroll
            for (int i = 1; i < A_DIM; ++i) mx = fmaxf(mx, row[i]);
            float Z = 0.0f, S1 = 0.0f;
#pragma unroll
            for (int i = 0; i < A_DIM; ++i) {
                float x = row[i] - mx;
                float e = expf(x);
                Z  += e;
                S1 += e * x;
            }
            float logZ = logf(Z);
            float lse  = mx + logZ;
            int act = actions[t];
            act = act < 0 ? 0 : (act > A_DIM - 1 ? A_DIM - 1 : act);
            float nlp    = row[act] - lse;
            float entrow = logZ - S1 / Z;                 // -sum p*logp
            float ratio  = expf(nlp - old_log_probs[t]);
            float advn   = (adv[t] - meanv) * rstd;
            float s1 = -ratio * advn;
            float rc = fminf(fmaxf(ratio, 1.0f - CLAMP_), 1.0f + CLAMP_);
            float s2 = -rc * advn;
            pl += fmaxf(s1, s2);
            float dv = ret[t] - row[A_DIM];               // col 17 = state value
            vl += dv * dv;
            ent += entrow;
        }
        __syncthreads();
    }

    // Block reduction of the three partial sums -> 3 global float atomics
    red[tid] = pl; __syncthreads();
    for (int off = G5_THREADS / 2; off > 0; off >>= 1) {
        if (tid < off) red[tid] += red[tid + off];
        __syncthreads();
    }
    if (tid == 0) atomicAdd(&acc[2], red[0]);
    __syncthreads();
    red[tid] = vl; __syncthreads();
    for (int off = G5_THREADS / 2; off > 0; off >>= 1) {
        if (tid < off) red[tid] += red[tid + off];
        __syncthreads();
    }
    if (tid == 0) atomicAdd(&acc[3], red[0]);
    __syncthreads();
    red[tid] = ent; __syncthreads();
    for (int off = G5_THREADS / 2; off > 0; off >>= 1) {
        if (tid < off) red[tid] += red[tid + off];
        __syncthreads();
    }
    if (tid == 0) atomicAdd(&acc[4], red[0]);
}

// K6: combine into scalar loss
__global__ void finalize(const float* __restrict__ acc, float* __restrict__ out) {
    if (threadIdx.x == 0 && blockIdx.x == 0) {
        const float invT = 1.0f / (float)T_TOT;
        out[0] = acc[2] * invT + VALC_ * acc[3] * invT - ENTC_ * acc[4] * invT;
    }
}

// ------------------------------------------------------------------
extern "C" void kernel_launch(void* const* d_in, const int* in_sizes, int n_in,
                              void* d_out, int out_size, void* d_ws, size_t ws_size,
                              hipStream_t stream) {
    (void)in_sizes; (void)n_in; (void)out_size; (void)ws_size;
    const float* states        = (const float*)d_in[0];
    const int*   actions       = (const int*)  d_in[1];
    const float* old_log_probs = (const float*)d_in[2];
    const float* old_values    = (const float*)d_in[3];
    const float* rewards       = (const float*)d_in[4];
    const int*   terminated    = (const int*)  d_in[5];
    const float* W             = (const float*)d_in[6];
    const float* w_v           = (const float*)d_in[7];
    float* out = (float*)d_out;

    float* wsf   = (float*)d_ws;
    float* acc   = wsf;                  // [0]=sum [1]=sumsq [2]=pl [3]=vl [4]=ent [5]=mean [6]=rstd
    float* adv   = wsf + 16;
    float* ret   = adv + T_TOT;
    float* chA   = ret + T_TOT;
    float* chB   = chA + NCHUNK;
    float* carry = chB + NCHUNK;

    init_acc    <<<1, 32, 0, stream>>>(acc);
    gae_partials<<<NCHUNK, CTH, 0, stream>>>(rewards, old_values, terminated, chA, chB);
    gae_carry   <<<1, 32, 0, stream>>>(chA, chB, carry);
    gae_apply   <<<NCHUNK, CTH, 0, stream>>>(rewards, old_values, terminated, carry, adv, ret, acc);
    gae_stats   <<<1, 32, 0, stream>>>(acc);

    size_t smem = (size_t)(32 * WTS + WAVES_PER_BLK * 16 * STG + G5_THREADS) * sizeof(float);
    ppo_main    <<<G5_BLOCKS, G5_THREADS, smem, stream>>>(states, actions, old_log_probs,
                                                          W, w_v, adv, ret, acc);
    finalize    <<<1, 32, 0, stream>>>(acc, out);
}